// BatchLabelPropagation_86303072846485
// MI455X (gfx1250) — compile-verified
//
#include <hip/hip_runtime.h>
#include <hip/hip_bf16.h>
#include <math.h>

typedef __attribute__((ext_vector_type(2))) float v2f;
typedef __attribute__((ext_vector_type(8))) float v8f;

#define BATCH 16
#define E 128
#define CDIM 512
#define NCLS 5
#define ALPHA 0.2f

// ---------------------------------------------------------------------------
// 16x16 tile GEMM helper built on V_WMMA_F32_16X16X4_F32.
// A-frag layout (16x4 f32): lanes 0-15 rows M=0..15 K={0,1}; lanes 16-31 K={2,3}.
// B-frag layout (4x16 f32): VGPR0 lanes0-15 = (K=kb, N=lane), VGPR1 = K=kb+1.
// C/D (16x16 f32, 8 VGPRs): VGPR r: lanes0-15 -> M=r, lanes16-31 -> M=r+8.
// TB=true reads B transposed (B stored row-major as N x K), i.e. C = A * B^T.
// ---------------------------------------------------------------------------
template <bool TB>
__device__ inline v8f gemm_tile(const float* __restrict__ Al,
                                const float* __restrict__ Bl,
                                int rowBase, int colBase, int K,
                                int lda, int ldb, int lane) {
  const int m  = lane & 15;
  const int kb = (lane >> 4) << 1;
  v8f acc = {0.f, 0.f, 0.f, 0.f, 0.f, 0.f, 0.f, 0.f};
  for (int kk = 0; kk < K; kk += 4) {
    v2f a = *(const v2f*)&Al[(rowBase + m) * lda + kk + kb];
    v2f b;
    if (TB) {
      b = *(const v2f*)&Bl[(colBase + m) * ldb + kk + kb];
    } else {
      b.x = Bl[(kk + kb) * ldb + colBase + m];
      b.y = Bl[(kk + kb + 1) * ldb + colBase + m];
    }
    acc = __builtin_amdgcn_wmma_f32_16x16x4_f32(
        /*neg_a=*/false, a, /*neg_b=*/false, b,
        /*c_mod=*/(short)0, acc, /*reuse_a=*/false, /*reuse_b=*/false);
  }
  return acc;
}

__device__ inline void store_tile(float* __restrict__ Cm, int rowBase,
                                  int colBase, int lane, v8f acc) {
  const int n    = colBase + (lane & 15);
  const int mOff = rowBase + ((lane >> 4) << 3);
#pragma unroll
  for (int r = 0; r < 8; ++r) Cm[(mOff + r) * E + n] = acc[r];
}

// ---------------------------------------------------------------------------
// Kernel 1: per-batch Gram matrix via WMMA (NT), then squared distances.
// LDS: x slice (128x512 f32 = 256KB) + norms (128 f32). Fits in 320KB WGP LDS.
// ---------------------------------------------------------------------------
__global__ __launch_bounds__(256) void k_gram(const float* __restrict__ x,
                                              float* __restrict__ sqd) {
  extern __shared__ float smem[];
  float* xs    = smem;             // E * CDIM
  float* norms = smem + E * CDIM;  // E
  const int b    = blockIdx.x;
  const int tid  = threadIdx.x;
  const int lane = tid & 31;
  const int wave = tid >> 5;

  // Stage x[b] into LDS with float4 loads.
  const float4* xg  = (const float4*)(x + (size_t)b * E * CDIM);
  float4*       xs4 = (float4*)xs;
  for (int i = tid; i < E * CDIM / 4; i += 256) xs4[i] = xg[i];
  __syncthreads();

  // Row norms ||x_i||^2.
  for (int i = tid; i < E; i += 256) {
    const float* row = xs + i * CDIM;
    float s = 0.f;
    for (int k = 0; k < CDIM; ++k) s += row[k] * row[k];
    norms[i] = s;
  }
  __syncthreads();

  const float invs = 0.04419417382415922f;  // 1/sqrt(512)
  float* out = sqd + (size_t)b * E * E;
  // Wave w owns the row band [w*16, w*16+16); 8 column tiles each.
  for (int t = 0; t < 8; ++t) {
    v8f acc = gemm_tile<true>(xs, xs, wave * 16, t * 16, CDIM, CDIM, CDIM, lane);
    const int n    = t * 16 + (lane & 15);
    const int mOff = wave * 16 + ((lane >> 4) << 3);
#pragma unroll
    for (int r = 0; r < 8; ++r) {
      const int i = mOff + r;
      float s = (i == n) ? 0.f
                         : (norms[i] + norms[n] - 2.f * acc[r]) * invs;
      out[i * E + n] = s;
    }
  }
}

// ---------------------------------------------------------------------------
// Kernel 2: deterministic global masked mean/std over all batches.
// var uses Bessel correction (ddof=1), matching torch .std() on masked elems.
// Writes stats[0] = 1/sqrt(var).
// ---------------------------------------------------------------------------
__global__ __launch_bounds__(256) void k_stats(const float* __restrict__ sqd,
                                               float* __restrict__ stats) {
  __shared__ float s_sum[256], s_sq[256], s_cnt[256];
  const int tid = threadIdx.x;
  const int N   = BATCH * E * E;
  float sum = 0.f, sq = 0.f, cnt = 0.f;
  for (int i = tid; i < N; i += 256) {
    float v = sqd[i];
    sum += v;
    sq  += v * v;
    cnt += (v != 0.f) ? 1.f : 0.f;
  }
  s_sum[tid] = sum; s_sq[tid] = sq; s_cnt[tid] = cnt;
  __syncthreads();
  for (int off = 128; off > 0; off >>= 1) {
    if (tid < off) {
      s_sum[tid] += s_sum[tid + off];
      s_sq[tid]  += s_sq[tid + off];
      s_cnt[tid] += s_cnt[tid + off];
    }
    __syncthreads();
  }
  if (tid == 0) {
    float c    = s_cnt[0];
    float mean = s_sum[0] / c;
    float var  = (s_sq[0] - c * mean * mean) / (c - 1.f);
    stats[0]   = rsqrtf(var);
  }
}

// ---------------------------------------------------------------------------
// Kernel 3: per-batch weights -> A = I - alpha*S -> inverse via Newton-Schulz
// (X <- 2X - X(AX), 5 iters, all WMMA GEMMs in LDS) -> L1 norm -> label prop.
// LDS: A, X, T (3 x 128x128 f32) + rowsum + labels ~ 193KB.
// ---------------------------------------------------------------------------
__global__ __launch_bounds__(256) void k_solve(const float* __restrict__ sqd,
                                               const float* __restrict__ stats,
                                               const int* __restrict__ labels,
                                               float* __restrict__ out) {
  extern __shared__ float smem[];
  float* Amat = smem;              // E*E
  float* X    = smem + E * E;      // E*E
  float* T    = smem + 2 * E * E;  // E*E
  float* red  = smem + 3 * E * E;  // E
  int*   lab  = (int*)(smem + 3 * E * E + E);  // E
  const int b    = blockIdx.x;
  const int tid  = threadIdx.x;
  const int lane = tid & 31;
  const int wave = tid >> 5;
  const float rstd = stats[0];
  const float* sq  = sqd + (size_t)b * E * E;

  if (tid < E) lab[tid] = labels[b * E + tid];

  // weights (RBF, zero diagonal) into X
  for (int idx = tid; idx < E * E; idx += 256) {
    const int i = idx >> 7, j = idx & 127;
    X[idx] = (i == j) ? 0.f : expf(-sq[idx] * rstd);
  }
  __syncthreads();

  // red[j] = alpha / (1e-4 + rowsum_j)  (reference applies column factor twice)
  if (tid < E) {
    float r = 0.f;
    for (int j = 0; j < E; ++j) r += X[tid * E + j];
    red[tid] = ALPHA / (1e-4f + r);
  }
  __syncthreads();

  // A = I - alpha*S ; then X = I (Newton-Schulz seed; rho(alpha*S) < 0.2)
  for (int idx = tid; idx < E * E; idx += 256) {
    const int i = idx >> 7, j = idx & 127;
    Amat[idx] = ((i == j) ? 1.f : 0.f) - X[idx] * red[j];
  }
  __syncthreads();
  for (int idx = tid; idx < E * E; idx += 256) {
    const int i = idx >> 7, j = idx & 127;
    X[idx] = (i == j) ? 1.f : 0.f;
  }
  __syncthreads();

  v8f acc[8];
  for (int it = 0; it < 5; ++it) {
    // T = A @ X
    for (int t = 0; t < 8; ++t)
      acc[t] = gemm_tile<false>(Amat, X, wave * 16, t * 16, E, E, E, lane);
    for (int t = 0; t < 8; ++t) store_tile(T, wave * 16, t * 16, lane, acc[t]);
    __syncthreads();
    // U = X @ T (kept in regs)
    for (int t = 0; t < 8; ++t)
      acc[t] = gemm_tile<false>(X, T, wave * 16, t * 16, E, E, E, lane);
    __syncthreads();  // all reads of X complete before overwrite
    // X = 2X - U
    for (int t = 0; t < 8; ++t) {
      const int n    = t * 16 + (lane & 15);
      const int mOff = wave * 16 + ((lane >> 4) << 3);
#pragma unroll
      for (int r = 0; r < 8; ++r) {
        const int i = mOff + r;
        X[i * E + n] = 2.f * X[i * E + n] - acc[t][r];
      }
    }
    __syncthreads();
  }

  // L1 row-normalize + one-hot label propagation (drop class==NCLS) + log
  if (tid < E) {
    const int i = tid;
    float cls[NCLS] = {0.f, 0.f, 0.f, 0.f, 0.f};
    float l1 = 0.f;
    for (int j = 0; j < E; ++j) {
      const float p = X[i * E + j];
      l1 += fabsf(p);
      const int lb = lab[j];
      if (lb < NCLS) cls[lb] += p;
    }
    const float inv = 1.f / fmaxf(l1, 1e-12f);
#pragma unroll
    for (int c = 0; c < NCLS; ++c)
      out[((size_t)b * E + i) * NCLS + c] = logf(cls[c] * inv + 1e-6f);
  }
}

extern "C" void kernel_launch(void* const* d_in, const int* in_sizes, int n_in,
                              void* d_out, int out_size, void* d_ws, size_t ws_size,
                              hipStream_t stream) {
  const float* x      = (const float*)d_in[0];
  const int*   labels = (const int*)d_in[1];
  // d_in[2] = nclasses scalar (compile-time NCLS=5 here)
  float* out   = (float*)d_out;
  float* sqd   = (float*)d_ws;             // BATCH*E*E floats = 1 MB
  float* stats = sqd + BATCH * E * E;      // 1 float

  const size_t sh1 = (size_t)(E * CDIM + E) * sizeof(float);
  k_gram<<<BATCH, 256, sh1, stream>>>(x, sqd);
  k_stats<<<1, 256, 0, stream>>>(sqd, stats);
  const size_t sh3 = (size_t)(3 * E * E + 2 * E) * sizeof(float);
  k_solve<<<BATCH, 256, sh3, stream>>>(sqd, stats, labels, out);
}